// GATBase_4784593568417
// MI455X (gfx1250) — compile-verified
//
#include <hip/hip_runtime.h>
#include <math.h>

#define NN 30000
#define EE 480000
#define DD 128
#define HH 4
#define CCH 32
#define LL 4
#define D3 384
#define ECHUNK 96000   // divides EE (5 chunks), multiple of 128

typedef __attribute__((ext_vector_type(2))) float v2f;
typedef __attribute__((ext_vector_type(8))) float v8f;
typedef __attribute__((ext_vector_type(4))) float f4;

// ---------------------------------------------------------------- atomics
__device__ __forceinline__ void atomicMaxF(float* addr, float val) {
    int* ai = (int*)addr;
    int old = __float_as_int(*addr);
    while (__int_as_float(old) < val) {
        int assumed = old;
        old = atomicCAS(ai, assumed, __float_as_int(val));
        if (old == assumed) break;
    }
}

// ---------------------------------------------------------------- WMMA tiles
#define BM 128
#define BN 64
#define BK 32
#define ASTR 36    // A LDS stride (dwords): %4==0 for f4 stores, odd/64 bank spread
#define BPSTR 160  // B LDS stride per k-PAIR (dwords): == 32 mod 64 so the two
                   // lane halves (pair p vs p+1) hit disjoint 32-bank windows

// Stage one BM x BK A-chunk (optionally gathered) and one BK x BN B-chunk
// (k-pair interleaved so b-fragments are single ds_load_b64), then run the
// 16x16x4 f32 WMMA ladder, accumulating into acc[4].
__device__ __forceinline__ void wmma_chunk(
    float* As, float* Bs,
    const float* __restrict__ A, const int* __restrict__ aidx,
    const float* __restrict__ B,
    int M, int N, int K, int m0, int n0, int k0,
    int tid, int mloc, int khalf, int ncol, v8f acc[4])
{
    #pragma unroll
    for (int it = 0; it < 4; ++it) {
        int idx = tid + it * 256;
        int r   = idx >> 3;          // 0..127
        int c4  = (idx & 7) << 2;    // 0,4,..,28
        f4 v = {0.f, 0.f, 0.f, 0.f};
        int grow = m0 + r;
        if (grow < M) {
            int ar = aidx ? aidx[grow] : grow;
            v = *(const f4*)(A + (size_t)ar * K + k0 + c4);
        }
        *(f4*)&As[r * ASTR + c4] = v;
    }
    {
        // B chunk 32x64 -> Bs[kp*BPSTR + 2*n + (k&1)], kp = k>>1
        int kp = tid >> 4;            // 0..15
        int c4 = (tid & 15) << 2;     // 0,4,..,60
        f4 e = *(const f4*)(B + (size_t)(k0 + 2 * kp)     * N + n0 + c4);
        f4 o = *(const f4*)(B + (size_t)(k0 + 2 * kp + 1) * N + n0 + c4);
        f4 lo = {e.x, o.x, e.y, o.y};
        f4 hi = {e.z, o.z, e.w, o.w};
        *(f4*)&Bs[kp * BPSTR + 2 * c4]     = lo;
        *(f4*)&Bs[kp * BPSTR + 2 * c4 + 4] = hi;
    }
    __syncthreads();

    #pragma unroll
    for (int kk = 0; kk < BK; kk += 4) {
        v2f a;
        a.x = As[mloc * ASTR + kk + khalf];
        a.y = As[mloc * ASTR + kk + khalf + 1];
        int bbase = ((kk >> 1) + (khalf >> 1)) * BPSTR;  // pair index * stride
        #pragma unroll
        for (int nt = 0; nt < 4; ++nt) {
            v2f b = *(const v2f*)&Bs[bbase + 2 * (nt * 16 + ncol)];
            acc[nt] = __builtin_amdgcn_wmma_f32_16x16x4_f32(
                false, a, false, b, (short)0, acc[nt], false, false);
        }
    }
    __syncthreads();
}

// C[M,N] = A[M,K] @ B[K,N] (+ bias[n]) (+ C if cmode==1), row-major fp32.
// Optional row gather via aidx. Requires K%32==0, N%64==0; M ragged ok.
__global__ __launch_bounds__(256) void gemm_f32_wmma(
    const float* __restrict__ A, const int* __restrict__ aidx,
    const float* __restrict__ B, float* __restrict__ C,
    const float* __restrict__ bias,
    int M, int N, int K, int cmode)
{
    __shared__ float As[BM * ASTR];
    __shared__ float Bs[(BK / 2) * BPSTR];

    const int tid  = threadIdx.x;
    const int wave = tid >> 5;
    const int lane = tid & 31;
    const int m0 = blockIdx.x * BM;
    const int n0 = blockIdx.y * BN;

    const int mloc  = wave * 16 + (lane & 15);
    const int khalf = (lane >> 4) * 2;
    const int ncol  = lane & 15;
    const int rbase = (lane >> 4) * 8;

    v8f acc[4] = {};
    for (int k0 = 0; k0 < K; k0 += BK)
        wmma_chunk(As, Bs, A, aidx, B, M, N, K, m0, n0, k0,
                   tid, mloc, khalf, ncol, acc);

    for (int nt = 0; nt < 4; ++nt) {
        int cn = n0 + nt * 16 + ncol;
        float bv = bias ? bias[cn] : 0.f;
        for (int rr = 0; rr < 8; ++rr) {
            int r = m0 + wave * 16 + rr + rbase;
            if (r < M) {
                size_t o = (size_t)r * N + cn;
                float v = acc[nt][rr] + bv;
                if (cmode == 1) v += C[o];
                C[o] = v;
            }
        }
    }
}

// Fused xcat GEMM: out[M,384] = [nf[row]|nf[col]|ef] @ w1[384,384] + b1.
// Three K=128 segments iterate different A sources; tmp written exactly once.
__global__ __launch_bounds__(256) void gemm_xcat_wmma(
    const float* __restrict__ nf, const float* __restrict__ efc,
    const int* __restrict__ rowp, const int* __restrict__ colp,
    const float* __restrict__ w1, const float* __restrict__ b1,
    float* __restrict__ out, int M)
{
    __shared__ float As[BM * ASTR];
    __shared__ float Bs[(BK / 2) * BPSTR];

    const int tid  = threadIdx.x;
    const int wave = tid >> 5;
    const int lane = tid & 31;
    const int m0 = blockIdx.x * BM;
    const int n0 = blockIdx.y * BN;

    const int mloc  = wave * 16 + (lane & 15);
    const int khalf = (lane >> 4) * 2;
    const int ncol  = lane & 15;
    const int rbase = (lane >> 4) * 8;

    v8f acc[4] = {};
    #pragma unroll
    for (int seg = 0; seg < 3; ++seg) {
        const float* Aseg = (seg == 2) ? efc : nf;
        const int*   idx  = (seg == 0) ? rowp : (seg == 1) ? colp : nullptr;
        const float* Bseg = w1 + (size_t)seg * DD * D3;
        for (int k0 = 0; k0 < DD; k0 += BK)
            wmma_chunk(As, Bs, Aseg, idx, Bseg, M, D3, DD, m0, n0, k0,
                       tid, mloc, khalf, ncol, acc);
    }

    for (int nt = 0; nt < 4; ++nt) {
        int cn = n0 + nt * 16 + ncol;
        float bv = b1[cn];
        for (int rr = 0; rr < 8; ++rr) {
            int r = m0 + wave * 16 + rr + rbase;
            if (r < M) out[(size_t)r * D3 + cn] = acc[nt][rr] + bv;
        }
    }
}

// ---------------------------------------------------------------- encoders
__global__ void embed_nodes(const int* __restrict__ x,
                            const float* __restrict__ emb, float* __restrict__ nf) {
    int t = blockIdx.x * 256 + threadIdx.x;
    if (t >= NN * DD) return;
    int n = t >> 7, d = t & 127;
    float s = 0.f;
    #pragma unroll
    for (int i = 0; i < 9; ++i)
        s += emb[((size_t)i * 100 + x[n * 9 + i]) * DD + d];
    nf[t] = s;
}

__global__ void embed_edges(const int* __restrict__ ea,
                            const float* __restrict__ emb, float* __restrict__ ef) {
    int t = blockIdx.x * 256 + threadIdx.x;
    if (t >= EE * DD) return;
    int e = t >> 7, d = t & 127;
    float s = 0.f;
    #pragma unroll
    for (int i = 0; i < 3; ++i)
        s += emb[((size_t)i * 100 + ea[e * 3 + i]) * DD + d];
    ef[t] = s;
}

// ---------------------------------------------------------------- GAT pieces
// wea[d][h] = sum_c lin_edge_w[d, h*32+c] * a_edge[h,c]   (folds ep.a_edge)
__global__ void compute_wea(const float* __restrict__ lew,
                            const float* __restrict__ aed, float* __restrict__ wea) {
    int t = blockIdx.x * 128 + threadIdx.x;
    if (t >= DD * HH) return;
    int d = t >> 2, h = t & 3;
    float s = 0.f;
    for (int c = 0; c < CCH; ++c) s += lew[d * DD + h * CCH + c] * aed[h * CCH + c];
    wea[t] = s;
}

__global__ void node_att(const float* __restrict__ xs, const float* __restrict__ asr,
                         const float* __restrict__ ads, float* __restrict__ ssrc,
                         float* __restrict__ sdst) {
    int t = blockIdx.x * 256 + threadIdx.x;
    if (t >= NN * HH) return;
    int n = t >> 2, h = t & 3;
    const float* xr = xs + (size_t)n * DD + h * CCH;
    float s1 = 0.f, s2 = 0.f;
    for (int c = 0; c < CCH; ++c) { s1 += xr[c] * asr[h * CCH + c]; s2 += xr[c] * ads[h * CCH + c]; }
    ssrc[t] = s1; sdst[t] = s2;
}

__global__ void gat_init(float* __restrict__ maxb, float* __restrict__ sumb,
                         float* __restrict__ conv) {
    int t = blockIdx.x * 256 + threadIdx.x;
    if (t < NN * HH) { maxb[t] = -3.0e38f; sumb[t] = 0.f; }
    if (t < NN * DD) conv[t] = 0.f;
}

__global__ __launch_bounds__(256) void edge_logits(
    const float* __restrict__ ef, const float* __restrict__ wea,
    const float* __restrict__ ssrc, const float* __restrict__ sdst,
    const int* __restrict__ row, const int* __restrict__ col,
    float* __restrict__ elog, float* __restrict__ maxb)
{
    __shared__ float sw[DD * HH];
    for (int i = threadIdx.x; i < DD * HH; i += 256) sw[i] = wea[i];
    __syncthreads();
    int e = blockIdx.x * 256 + threadIdx.x;
    if (e >= EE) return;
    const f4* efr = (const f4*)(ef + (size_t)e * DD);
    float acc[4] = {0.f, 0.f, 0.f, 0.f};
    for (int q = 0; q < 32; ++q) {
        f4 v = efr[q];
        int d = q * 4;
        #pragma unroll
        for (int j = 0; j < 4; ++j)
            #pragma unroll
            for (int h = 0; h < 4; ++h) acc[h] += v[j] * sw[(d + j) * 4 + h];
    }
    int r = row[e], c = col[e];
    #pragma unroll
    for (int h = 0; h < 4; ++h) {
        float lg = ssrc[r * 4 + h] + sdst[c * 4 + h] + acc[h];
        lg = lg > 0.f ? lg : 0.2f * lg;
        elog[e * 4 + h] = lg;
        atomicMaxF(&maxb[c * 4 + h], lg);
    }
}

__global__ void self_logits(const float* __restrict__ ssrc, const float* __restrict__ sdst,
                            float* __restrict__ slog, float* __restrict__ maxb) {
    int t = blockIdx.x * 256 + threadIdx.x;
    if (t >= NN * HH) return;
    float lg = ssrc[t] + sdst[t];
    lg = lg > 0.f ? lg : 0.2f * lg;
    slog[t] = lg;
    atomicMaxF(&maxb[t], lg);
}

__global__ void edge_expsum(const int* __restrict__ col, float* __restrict__ elog,
                            const float* __restrict__ maxb, float* __restrict__ sumb) {
    int t = blockIdx.x * 256 + threadIdx.x;
    if (t >= EE * HH) return;
    int e = t >> 2, h = t & 3;
    int c = col[e];
    float v = expf(elog[t] - maxb[c * 4 + h]);
    elog[t] = v;
    atomicAdd(&sumb[c * 4 + h], v);
}

__global__ void self_expsum(float* __restrict__ slog, const float* __restrict__ maxb,
                            float* __restrict__ sumb) {
    int t = blockIdx.x * 256 + threadIdx.x;
    if (t >= NN * HH) return;
    float v = expf(slog[t] - maxb[t]);
    slog[t] = v;
    atomicAdd(&sumb[t], v);
}

__global__ __launch_bounds__(256) void edge_scatter(
    const int* __restrict__ row, const int* __restrict__ col,
    const float* __restrict__ elog, const float* __restrict__ sumb,
    const float* __restrict__ xs, float* __restrict__ conv)
{
    int e = blockIdx.x * 8 + (threadIdx.x >> 5);
    if (e >= EE) return;
    int lane = threadIdx.x & 31;
    int r = row[e], c = col[e];
    int h = lane >> 3;
    float a = elog[e * 4 + h] / (sumb[c * 4 + h] + 1e-16f);
    f4 xv = ((const f4*)(xs + (size_t)r * DD))[lane];
    float* dst = conv + (size_t)c * DD + lane * 4;
    atomicAdd(dst + 0, xv.x * a);
    atomicAdd(dst + 1, xv.y * a);
    atomicAdd(dst + 2, xv.z * a);
    atomicAdd(dst + 3, xv.w * a);
}

__global__ void self_scatter(const float* __restrict__ slog, const float* __restrict__ sumb,
                             const float* __restrict__ xs, float* __restrict__ conv) {
    int t = blockIdx.x * 256 + threadIdx.x;
    if (t >= NN * DD) return;
    int n = t >> 7, d = t & 127, h = d >> 5;
    float a = slog[n * 4 + h] / (sumb[n * 4 + h] + 1e-16f);
    conv[t] += xs[t] * a;
}

// nf = relu(LN(conv + gat_bias)) + nf    (one wave per node, D=128)
__global__ __launch_bounds__(256) void node_norm(
    const float* __restrict__ conv, const float* __restrict__ bias,
    const float* __restrict__ lns, const float* __restrict__ lnb,
    float* __restrict__ nf)
{
    int n = blockIdx.x * 8 + (threadIdx.x >> 5);
    if (n >= NN) return;
    int lane = threadIdx.x & 31;
    f4 v = ((const f4*)(conv + (size_t)n * DD))[lane];
    f4 bb = ((const f4*)bias)[lane];
    v += bb;
    float s  = v.x + v.y + v.z + v.w;
    float ss = v.x * v.x + v.y * v.y + v.z * v.z + v.w * v.w;
    for (int o = 16; o > 0; o >>= 1) { s += __shfl_xor(s, o); ss += __shfl_xor(ss, o); }
    float mean = s * (1.f / 128.f);
    float var  = ss * (1.f / 128.f) - mean * mean;
    float rstd = rsqrtf(var + 1e-5f);
    f4 sc = ((const f4*)lns)[lane];
    f4 bt = ((const f4*)lnb)[lane];
    float* op = nf + (size_t)n * DD + lane * 4;
    f4 old = *(const f4*)op;
    f4 y;
    y.x = fmaxf((v.x - mean) * rstd * sc.x + bt.x, 0.f) + old.x;
    y.y = fmaxf((v.y - mean) * rstd * sc.y + bt.y, 0.f) + old.y;
    y.z = fmaxf((v.z - mean) * rstd * sc.z + bt.z, 0.f) + old.z;
    y.w = fmaxf((v.w - mean) * rstd * sc.w + bt.w, 0.f) + old.w;
    *(f4*)op = y;
}

// h = relu(LN(tmp1)) in place (bias already fused into GEMM epilogue),
// per edge row of 384 (one wave per edge)
__global__ __launch_bounds__(256) void edge_ln_relu(
    float* __restrict__ tmp1,
    const float* __restrict__ lns, const float* __restrict__ lnb, int count)
{
    int e = blockIdx.x * 8 + (threadIdx.x >> 5);
    if (e >= count) return;
    int lane = threadIdx.x & 31;
    float* rowp = tmp1 + (size_t)e * D3;
    f4 v[3];
    float s = 0.f, ss = 0.f;
    #pragma unroll
    for (int k = 0; k < 3; ++k) {
        f4 t = ((const f4*)rowp)[lane + 32 * k];
        v[k] = t;
        s  += t.x + t.y + t.z + t.w;
        ss += t.x * t.x + t.y * t.y + t.z * t.z + t.w * t.w;
    }
    for (int o = 16; o > 0; o >>= 1) { s += __shfl_xor(s, o); ss += __shfl_xor(ss, o); }
    float mean = s * (1.f / 384.f);
    float var  = ss * (1.f / 384.f) - mean * mean;
    float rstd = rsqrtf(var + 1e-5f);
    #pragma unroll
    for (int k = 0; k < 3; ++k) {
        f4 sc = ((const f4*)lns)[lane + 32 * k];
        f4 bt = ((const f4*)lnb)[lane + 32 * k];
        f4 y;
        y.x = fmaxf((v[k].x - mean) * rstd * sc.x + bt.x, 0.f);
        y.y = fmaxf((v[k].y - mean) * rstd * sc.y + bt.y, 0.f);
        y.z = fmaxf((v[k].z - mean) * rstd * sc.z + bt.z, 0.f);
        y.w = fmaxf((v[k].w - mean) * rstd * sc.w + bt.w, 0.f);
        ((f4*)rowp)[lane + 32 * k] = y;
    }
}

// ---------------------------------------------------------------- launch
extern "C" void kernel_launch(void* const* d_in, const int* in_sizes, int n_in,
                              void* d_out, int out_size, void* d_ws, size_t ws_size,
                              hipStream_t stream) {
    const int*   x        = (const int*)d_in[0];
    const int*   eattr    = (const int*)d_in[1];
    const int*   eidx     = (const int*)d_in[2];
    const float* atom_emb = (const float*)d_in[3];
    const float* bond_emb = (const float*)d_in[4];
    const float* lin_w    = (const float*)d_in[5];
    const float* att_src  = (const float*)d_in[6];
    const float* att_dst  = (const float*)d_in[7];
    const float* att_edge = (const float*)d_in[8];
    const float* lin_ew   = (const float*)d_in[9];
    const float* gat_bias = (const float*)d_in[10];
    const float* ln_s     = (const float*)d_in[11];
    const float* ln_b     = (const float*)d_in[12];
    const float* eu_w1    = (const float*)d_in[13];
    const float* eu_b1    = (const float*)d_in[14];
    const float* eu_lns   = (const float*)d_in[15];
    const float* eu_lnb   = (const float*)d_in[16];
    const float* eu_w2    = (const float*)d_in[17];
    const float* eu_b2    = (const float*)d_in[18];

    const int* row = eidx;
    const int* col = eidx + EE;

    float* nf = (float*)d_out;
    float* ef = nf + (size_t)NN * DD;

    float* w = (float*)d_ws;
    float* xs   = w;  w += (size_t)NN * DD;
    float* conv = w;  w += (size_t)NN * DD;
    float* ssrc = w;  w += (size_t)NN * HH;
    float* sdst = w;  w += (size_t)NN * HH;
    float* maxb = w;  w += (size_t)NN * HH;
    float* sumb = w;  w += (size_t)NN * HH;
    float* slog = w;  w += (size_t)NN * HH;
    float* elog = w;  w += (size_t)EE * HH;
    float* wea  = w;  w += (size_t)DD * HH;
    float* tmp1 = w;  w += (size_t)ECHUNK * D3;

    embed_nodes<<<(NN * DD + 255) / 256, 256, 0, stream>>>(x, atom_emb, nf);
    embed_edges<<<(EE * DD + 255) / 256, 256, 0, stream>>>(eattr, bond_emb, ef);

    for (int l = 0; l < LL; ++l) {
        const float* lw  = lin_w    + (size_t)l * DD * DD;
        const float* asr = att_src  + (size_t)l * HH * CCH;
        const float* ads = att_dst  + (size_t)l * HH * CCH;
        const float* aed = att_edge + (size_t)l * HH * CCH;
        const float* lew = lin_ew   + (size_t)l * DD * DD;
        const float* gb  = gat_bias + (size_t)l * DD;
        const float* ls  = ln_s     + (size_t)l * DD;
        const float* lb  = ln_b     + (size_t)l * DD;
        const float* w1  = eu_w1    + (size_t)l * D3 * D3;
        const float* b1  = eu_b1    + (size_t)l * D3;
        const float* els = eu_lns   + (size_t)l * D3;
        const float* elb = eu_lnb   + (size_t)l * D3;
        const float* w2  = eu_w2    + (size_t)l * D3 * DD;
        const float* b2  = eu_b2    + (size_t)l * DD;

        // ---- GAT conv ----
        compute_wea<<<(DD * HH + 127) / 128, 128, 0, stream>>>(lew, aed, wea);
        {
            dim3 g((NN + BM - 1) / BM, DD / BN);
            gemm_f32_wmma<<<g, 256, 0, stream>>>(nf, nullptr, lw, xs, nullptr,
                                                 NN, DD, DD, 0);
        }
        node_att<<<(NN * HH + 255) / 256, 256, 0, stream>>>(xs, asr, ads, ssrc, sdst);
        gat_init<<<(NN * DD + 255) / 256, 256, 0, stream>>>(maxb, sumb, conv);
        edge_logits<<<EE / 256, 256, 0, stream>>>(ef, wea, ssrc, sdst, row, col, elog, maxb);
        self_logits<<<(NN * HH + 255) / 256, 256, 0, stream>>>(ssrc, sdst, slog, maxb);
        edge_expsum<<<EE * HH / 256, 256, 0, stream>>>(col, elog, maxb, sumb);
        self_expsum<<<(NN * HH + 255) / 256, 256, 0, stream>>>(slog, maxb, sumb);
        edge_scatter<<<EE / 8, 256, 0, stream>>>(row, col, elog, sumb, xs, conv);
        self_scatter<<<(NN * DD + 255) / 256, 256, 0, stream>>>(slog, sumb, xs, conv);
        node_norm<<<(NN + 7) / 8, 256, 0, stream>>>(conv, gb, ls, lb, nf);

        // ---- edge update (chunked over E) ----
        for (int c0 = 0; c0 < EE; c0 += ECHUNK) {
            float* efc = ef + (size_t)c0 * DD;
            // tmp1 = [nf[row]|nf[col]|ef] @ w1 + b1   (single fused pass)
            dim3 g1(ECHUNK / BM, D3 / BN);
            gemm_xcat_wmma<<<g1, 256, 0, stream>>>(nf, efc, row + c0, col + c0,
                                                   w1, b1, tmp1, ECHUNK);
            // tmp1 = relu(LN(tmp1))
            edge_ln_relu<<<ECHUNK / 8, 256, 0, stream>>>(tmp1, els, elb, ECHUNK);
            // ef += tmp1 @ w2 + b2   (epilogue-fused, no tmp2)
            dim3 g2(ECHUNK / BM, DD / BN);
            gemm_f32_wmma<<<g2, 256, 0, stream>>>(tmp1, nullptr, w2, efc, b2,
                                                  ECHUNK, DD, D3, 1);
        }
    }
}